// StaticScaledRealQuantizer_90692529423058
// MI455X (gfx1250) — compile-verified
//
#include <hip/hip_runtime.h>

// ---------------------------------------------------------------------------
// AWQ uint4 dequantize: out[8192][8192] = (q - zp) * s, group_size = 128,
// nibble order INV[k] = (k>>1) + 4*(k&1)  (shifts 0,16,4,20,8,24,12,28).
//
// Memory-bound kernel (≈290 MiB traffic -> ~12.4 us floor @ 23.3 TB/s).
// CDNA5 path used: global_load_async_to_lds_b128 (ASYNCcnt) to stage the
// packed row, s_wait_asynccnt, LDS broadcast of per-group scale/zp*s,
// non-temporal b128 stores for the 256 MiB output stream.
// ---------------------------------------------------------------------------

#define OUT_DIM 8192
#define IN_DIM  8192
#define GS      128
#define NGROUP  (IN_DIM / GS)            // 64 groups per row
#define ZPW     (NGROUP / 8)             // 8 packed zp words per row
#define WPR     (IN_DIM / 8)             // 1024 packed int32 words per row
#define BLOCK   256                      // 8 wave32s; 4 words / 32 floats per thread

typedef __attribute__((ext_vector_type(4))) float fvec4;
typedef __attribute__((ext_vector_type(4))) int   ivec4;

// Low 32 bits of a flat pointer into the LDS aperture == LDS byte offset
// (ISA 10.2).
__device__ __forceinline__ unsigned lds_off32(const void* p) {
  return (unsigned)(unsigned long long)p;
}

__global__ __launch_bounds__(BLOCK) void awq_dequant_kernel(
    const int*   __restrict__ Xp,     // [OUT_DIM][WPR] packed uint4 (8/word)
    const float* __restrict__ scale,  // [NGROUP][OUT_DIM] (transposed)
    const int*   __restrict__ zpp,    // [OUT_DIM][ZPW] packed uint4 zero points
    float*       __restrict__ out)    // [OUT_DIM][IN_DIM]
{
  __shared__ int   s_words[WPR];      // 4 KiB: one packed row
  __shared__ float s_s [NGROUP];      // per-group scale
  __shared__ float s_zs[NGROUP];      // per-group zp*scale

  const int o = blockIdx.x;           // output row
  const int t = threadIdx.x;

  // ---- CDNA5 async copy: stage the packed row global -> LDS (16 B / lane) --
  {
    const int* gsrc = Xp + (size_t)o * WPR + 4 * t;
    unsigned   ldst = lds_off32(&s_words[4 * t]);
    asm volatile("global_load_async_to_lds_b128 %0, %1, off"
                 :: "v"(ldst), "v"(gsrc)
                 : "memory");
  }

  // ---- stage per-group scale and zp*scale while the async copy runs -------
  if (t < NGROUP) {
    const int   g   = t;
    const float sv  = scale[(size_t)g * OUT_DIM + o];
    const int   zpw = zpp[o * ZPW + (g >> 3)];
    const int   k   = g & 7;                         // nibble index within word
    const int   sh  = ((k >> 1) << 2) + ((k & 1) << 4); // 4*INV[k]
    const float zp  = (float)((zpw >> sh) & 0xF);
    s_s [g] = sv;
    s_zs[g] = zp * sv;
  }

  // ---- wait for our async transfers, then workgroup barrier ---------------
#if __has_builtin(__builtin_amdgcn_s_wait_asynccnt)
  __builtin_amdgcn_s_wait_asynccnt(0);
#else
  asm volatile("s_wait_asynccnt 0" ::: "memory");
#endif
  __syncthreads();

  // ---- dequantize 32 consecutive elements (one group slice) per thread ----
  const int   g  = t >> 2;            // 4 threads per 128-wide group
  const float s  = s_s [g];
  const float zs = s_zs[g];

  const ivec4 w4 = *reinterpret_cast<const ivec4*>(&s_words[4 * t]);

  float* orow = out + (size_t)o * IN_DIM + 32 * t;   // 128 B contiguous/thread

#pragma unroll
  for (int j = 0; j < 4; ++j) {
    const int w = w4[j];
    fvec4 lo, hi;
    // element k uses shift 4*INV[k]: 0,16,4,20, 8,24,12,28
    lo.x = fmaf((float)((w      ) & 0xF), s, -zs);
    lo.y = fmaf((float)((w >> 16) & 0xF), s, -zs);
    lo.z = fmaf((float)((w >>  4) & 0xF), s, -zs);
    lo.w = fmaf((float)((w >> 20) & 0xF), s, -zs);
    hi.x = fmaf((float)((w >>  8) & 0xF), s, -zs);
    hi.y = fmaf((float)((w >> 24) & 0xF), s, -zs);
    hi.z = fmaf((float)((w >> 12) & 0xF), s, -zs);
    hi.w = fmaf((float)((w >> 28) & 0xF), s, -zs);
    // 256 MiB streaming output: non-temporal so it doesn't thrash L2.
    __builtin_nontemporal_store(lo, reinterpret_cast<fvec4*>(orow + 8 * j));
    __builtin_nontemporal_store(hi, reinterpret_cast<fvec4*>(orow + 8 * j + 4));
  }
}

extern "C" void kernel_launch(void* const* d_in, const int* in_sizes, int n_in,
                              void* d_out, int out_size, void* d_ws, size_t ws_size,
                              hipStream_t stream) {
  (void)in_sizes; (void)n_in; (void)out_size; (void)d_ws; (void)ws_size;
  const int*   Xp    = (const int*)d_in[0];    // [8192][1024] int32
  const float* scale = (const float*)d_in[1];  // [64][8192]  float32
  const int*   zpp   = (const int*)d_in[2];    // [8192][8]   int32
  // d_in[3] = group_size scalar (128) — baked into the kernel constants.
  float* out = (float*)d_out;                  // [8192][8192] float32

  dim3 grid(OUT_DIM);
  dim3 block(BLOCK);
  hipLaunchKernelGGL(awq_dequant_kernel, grid, block, 0, stream,
                     Xp, scale, zpp, out);
}